// DictionaryLearning_49598282334715
// MI455X (gfx1250) — compile-verified
//
#include <hip/hip_runtime.h>
#include <stddef.h>

typedef __attribute__((ext_vector_type(2))) float v2f;
typedef __attribute__((ext_vector_type(8))) float v8f;

#define P_TOT 32768   // 8*64*64 patches
#define NDICT 512
#define CDIM  64
#define HWSZ  4096    // 64*64
#define KSP   5

// -------------------- zero fill (coeffs.T region) --------------------
__global__ void dl_zero_kernel(float* __restrict__ p, size_t n) {
  size_t i = (size_t)blockIdx.x * blockDim.x + threadIdx.x;
  const size_t stride = (size_t)gridDim.x * blockDim.x;
  for (; i < n; i += stride) p[i] = 0.0f;
}

// -------------------- G = D^T D + eps*I via V_WMMA_F32_16X16X4_F32 --------------------
// One wave -> one 16x16 tile of G. 512/16 = 32 tiles per dim -> 1024 waves.
__global__ void dl_gram_kernel(const float* __restrict__ D, float* __restrict__ G) {
  const int lane = threadIdx.x & 31;
  const int wid  = (blockIdx.x * blockDim.x + threadIdx.x) >> 5;
  const int i0 = (wid >> 5) * 16;
  const int j0 = (wid & 31) * 16;
  const int m  = lane & 15;
  const int kh = lane >> 4;  // K half-select per ISA A/B f32 layout
  v8f acc = {};
#pragma unroll
  for (int k0 = 0; k0 < CDIM; k0 += 4) {
    const int ka = k0 + 2 * kh;
    v2f a, b;
    a.x = D[ka * NDICT + i0 + m];         // A[m=i, k] = D[k, i]
    a.y = D[(ka + 1) * NDICT + i0 + m];
    b.x = D[ka * NDICT + j0 + m];         // B[k, n=j] = D[k, j]
    b.y = D[(ka + 1) * NDICT + j0 + m];
    acc = __builtin_amdgcn_wmma_f32_16x16x4_f32(false, a, false, b, (short)0, acc,
                                                false, false);
  }
#pragma unroll
  for (int vr = 0; vr < 8; ++vr) {
    const int row = i0 + kh * 8 + vr;   // C/D layout: lanes>=16 hold M=8..15
    const int col = j0 + m;
    float v = acc[vr];
    if (row == col) v += 1e-5f;         // DIAG_EPS
    G[row * NDICT + col] = v;
  }
}

// -------------------- h_bar = patches @ D via V_WMMA_F32_16X16X4_F32 --------------------
// One wave -> 16 patches x 16 atoms. 2048 * 32 = 65536 waves.
__global__ void dl_hbar_kernel(const float* __restrict__ z_e, const float* __restrict__ D,
                               float* __restrict__ hbar) {
  const int lane  = threadIdx.x & 31;
  const int wid   = (blockIdx.x * blockDim.x + threadIdx.x) >> 5;
  const int n0    = (wid & 31) * 16;
  const int pbase = (wid >> 5) * 16;
  const int b     = pbase >> 12;           // batch (4096 patches each; 16 | 4096)
  const int hwb   = pbase & (HWSZ - 1);
  const int m  = lane & 15;
  const int kh = lane >> 4;
  const float* zb = z_e + (size_t)b * (CDIM * HWSZ) + hwb + m;  // + c*HWSZ per channel
  v8f acc = {};
#pragma unroll
  for (int k0 = 0; k0 < CDIM; k0 += 4) {
    const int ka = k0 + 2 * kh;
    v2f a, bf;
    a.x  = zb[(size_t)ka * HWSZ];          // patches[pbase+m, ka]
    a.y  = zb[(size_t)(ka + 1) * HWSZ];
    bf.x = D[ka * NDICT + n0 + m];         // D[ka, n0+m]
    bf.y = D[(ka + 1) * NDICT + n0 + m];
    acc = __builtin_amdgcn_wmma_f32_16x16x4_f32(false, a, false, bf, (short)0, acc,
                                                false, false);
  }
  float* out = hbar + (size_t)pbase * NDICT + n0 + m;
#pragma unroll
  for (int vr = 0; vr < 8; ++vr)
    out[(size_t)(kh * 8 + vr) * NDICT] = acc[vr];
}

// -------------------- OMP helpers --------------------
// Select argmax of masked |h|, h = h_bar - sum_{j<K} x[j]*G[idx[j], :].
// First-index tie-break to match jnp.argmax.
template <int K>
__device__ __forceinline__ int dl_select(const float* __restrict__ G, const float hb[16],
                                         const int lane, const int* idx, const float* x) {
  float best = -1.0f;
  int bestn = 0x7FFFFFFF;
#pragma unroll
  for (int i = 0; i < 16; ++i) {
    const int n = i * 32 + lane;           // strided: coalesced G row loads
    float h = hb[i];
#pragma unroll
    for (int j = 0; j < K; ++j) h -= x[j] * G[idx[j] * NDICT + n];
    float s = fabsf(h);
#pragma unroll
    for (int j = 0; j < K; ++j)
      if (n == idx[j]) s = 0.0f;           // active mask
    if (s > best || (s == best && n < bestn)) { best = s; bestn = n; }
  }
#pragma unroll
  for (int off = 16; off >= 1; off >>= 1) {
    const float ob = __shfl_xor(best, off, 32);
    const int on   = __shfl_xor(bestn, off, 32);
    if (ob > best || (ob == best && on < bestn)) { best = ob; bestn = on; }
  }
  return bestn;
}

// Solve G[idx,idx] x = h_bar[idx] for NA<=5, fully unrolled so arrays stay in VGPRs.
template <int NA>
__device__ __forceinline__ void dl_solve(const float* __restrict__ G,
                                         const float* __restrict__ hrow,
                                         const int* idx, float* x) {
  float A[NA][NA], rhs[NA];
#pragma unroll
  for (int a = 0; a < NA; ++a) {
#pragma unroll
    for (int c = 0; c < NA; ++c) A[a][c] = G[idx[a] * NDICT + idx[c]];
    rhs[a] = hrow[idx[a]];
  }
#pragma unroll
  for (int a = 0; a < NA; ++a) {           // forward elimination (SPD, no pivot)
    const float inv = 1.0f / A[a][a];
#pragma unroll
    for (int r = a + 1; r < NA; ++r) {
      const float f = A[r][a] * inv;
#pragma unroll
      for (int c = a; c < NA; ++c) A[r][c] -= f * A[a][c];
      rhs[r] -= f * rhs[a];
    }
  }
#pragma unroll
  for (int a = NA - 1; a >= 0; --a) {      // back substitution
    float s = rhs[a];
#pragma unroll
    for (int c = a + 1; c < NA; ++c) s -= A[a][c] * x[c];
    x[a] = s / A[a][a];
  }
}

// -------------------- OMP: one wave per patch --------------------
__global__ void dl_omp_kernel(const float* __restrict__ z_e, const float* __restrict__ D,
                              const float* __restrict__ G, const float* __restrict__ hbar,
                              float* __restrict__ z_out, float* __restrict__ coeffsT,
                              float* __restrict__ partials) {
  const int lane = threadIdx.x & 31;
  const int p = (int)((blockIdx.x * blockDim.x + threadIdx.x) >> 5);
  const float* hrow = hbar + (size_t)p * NDICT;
  float hb[16];
#pragma unroll
  for (int i = 0; i < 16; ++i) hb[i] = hrow[i * 32 + lane];

  int idx[KSP];
  float x[KSP] = {0.f, 0.f, 0.f, 0.f, 0.f};

  idx[0] = dl_select<0>(G, hb, lane, idx, x);
  dl_solve<1>(G, hrow, idx, x);
  idx[1] = dl_select<1>(G, hb, lane, idx, x);
  dl_solve<2>(G, hrow, idx, x);
  idx[2] = dl_select<2>(G, hb, lane, idx, x);
  dl_solve<3>(G, hrow, idx, x);
  idx[3] = dl_select<3>(G, hb, lane, idx, x);
  dl_solve<4>(G, hrow, idx, x);
  idx[4] = dl_select<4>(G, hb, lane, idx, x);
  dl_solve<5>(G, hrow, idx, x);

  // recon[p, c] = sum_j x[j] * D[c, idx[j]]; 2 channels per lane
  float r0 = 0.0f, r1 = 0.0f;
#pragma unroll
  for (int j = 0; j < KSP; ++j) {
    r0 += x[j] * D[lane * NDICT + idx[j]];
    r1 += x[j] * D[(lane + 32) * NDICT + idx[j]];
  }
  const int b = p >> 12;
  const int hw = p & (HWSZ - 1);
  const size_t zi0 = (size_t)b * (CDIM * HWSZ) + (size_t)lane * HWSZ + hw;
  const size_t zi1 = zi0 + (size_t)32 * HWSZ;
  z_out[zi0] = r0;                          // z_dl_st == z_dl in forward value
  z_out[zi1] = r1;

  const float d0 = r0 - z_e[zi0];
  const float d1 = r1 - z_e[zi1];
  float err = d0 * d0 + d1 * d1;
#pragma unroll
  for (int off = 16; off >= 1; off >>= 1) err += __shfl_xor(err, off, 32);
  if (lane == 0) partials[p] = err;

  // scatter coeffs.T[idx[j], p] = x[j] (values replicated across lanes)
#pragma unroll
  for (int j = 0; j < KSP; ++j)
    if (lane == j) coeffsT[(size_t)idx[j] * P_TOT + p] = x[j];
}

// -------------------- deterministic loss reduction --------------------
__global__ void dl_loss_kernel(const float* __restrict__ partials, float* __restrict__ lossOut) {
  __shared__ float buf[256];
  float s = 0.0f;
  for (int i = threadIdx.x; i < P_TOT; i += 256) s += partials[i];
  buf[threadIdx.x] = s;
  __syncthreads();
  for (int o = 128; o >= 1; o >>= 1) {
    if ((int)threadIdx.x < o) buf[threadIdx.x] += buf[threadIdx.x + o];
    __syncthreads();
  }
  if (threadIdx.x == 0)
    lossOut[0] = 1.25f * buf[0] / 2097152.0f;  // (1 + COMMIT) * mean sq err
}

// -------------------- launcher --------------------
extern "C" void kernel_launch(void* const* d_in, const int* in_sizes, int n_in,
                              void* d_out, int out_size, void* d_ws, size_t ws_size,
                              hipStream_t stream) {
  const float* z_e = (const float*)d_in[0];  // [8,64,64,64]
  const float* D   = (const float*)d_in[1];  // [64,512]
  float* out      = (float*)d_out;
  float* z_out    = out;                     // 2097152 floats
  float* lossOut  = out + 2097152;           // 1 float
  float* coeffsT  = out + 2097153;           // 512*32768 floats

  float* ws       = (float*)d_ws;
  float* partials = ws;                      // 32768 floats
  float* G        = ws + P_TOT;              // 262144 floats
  float* hbar     = ws + P_TOT + NDICT * NDICT;  // 16777216 floats

  dl_zero_kernel<<<2048, 256, 0, stream>>>(coeffsT, (size_t)NDICT * P_TOT);
  dl_gram_kernel<<<128, 256, 0, stream>>>(D, G);          // 1024 waves
  dl_hbar_kernel<<<8192, 256, 0, stream>>>(z_e, D, hbar); // 65536 waves
  dl_omp_kernel<<<4096, 256, 0, stream>>>(z_e, D, G, hbar, z_out, coeffsT, partials);
  dl_loss_kernel<<<1, 256, 0, stream>>>(partials, lossOut);
}